// Encoder_LP_44109314130368
// MI455X (gfx1250) — compile-verified
//
#include <hip/hip_runtime.h>
#include <hip/hip_bf16.h>

#define N_NODES 100000
#define N_EDGES 1600000
#define DIM 64
#define OUT_DIM 32

typedef __attribute__((ext_vector_type(2))) float v2f;
typedef __attribute__((ext_vector_type(8))) float v8f;

// ---------------------------------------------------------------------------
// init: deg = 1.0 (self-loop), agg1 = agg2 = 0 (scatter-add targets)
// ---------------------------------------------------------------------------
__global__ void init_kernel(float* __restrict__ deg,
                            float* __restrict__ agg1,
                            float* __restrict__ agg2) {
    size_t tid = (size_t)blockIdx.x * blockDim.x + threadIdx.x;
    const size_t tot = (size_t)N_NODES * DIM;
    if (tid < tot) { agg1[tid] = 0.0f; agg2[tid] = 0.0f; }
    if (tid < N_NODES) deg[tid] = 1.0f;   // self-loop contribution
}

// deg[dst] += 1 over real edges
__global__ void degree_kernel(const int* __restrict__ dst, float* __restrict__ deg) {
    int e = blockIdx.x * blockDim.x + threadIdx.x;
    if (e < N_EDGES) atomicAdd(&deg[dst[e]], 1.0f);
}

__global__ void rsqrt_kernel(float* __restrict__ deg) {
    int i = blockIdx.x * blockDim.x + threadIdx.x;
    if (i < N_NODES) deg[i] = rsqrtf(deg[i]);
}

// ---------------------------------------------------------------------------
// GEMM1: xw[N,64] = x[N,64] @ W1[64,64], exact f32 via V_WMMA_F32_16X16X4_F32.
// One wave per 16-row block; 4 accumulators cover all 64 output columns.
// ---------------------------------------------------------------------------
__global__ __launch_bounds__(32) void gemm_x_w1(const float* __restrict__ x,
                                                const float* __restrict__ W,
                                                float* __restrict__ out) {
    const int lane = threadIdx.x;
    const int m    = lane & 15;
    const int half = lane >> 4;           // 0: K pair {0,1}, 1: K pair {2,3}
    const int row0 = blockIdx.x * 16;

    v8f acc0 = {}, acc1 = {}, acc2 = {}, acc3 = {};
    const float* xr = x + (size_t)(row0 + m) * DIM;

#pragma unroll
    for (int ks = 0; ks < 16; ++ks) {
        const int kk = ks * 4 + half * 2;
        v2f a;  a.x = xr[kk];  a.y = xr[kk + 1];
        const float* wr0 = W + (size_t)kk * DIM;
        const float* wr1 = W + (size_t)(kk + 1) * DIM;
        v2f b0, b1, b2, b3;
        b0.x = wr0[m];      b0.y = wr1[m];
        b1.x = wr0[16 + m]; b1.y = wr1[16 + m];
        b2.x = wr0[32 + m]; b2.y = wr1[32 + m];
        b3.x = wr0[48 + m]; b3.y = wr1[48 + m];
        acc0 = __builtin_amdgcn_wmma_f32_16x16x4_f32(false, a, false, b0, (short)0, acc0, false, false);
        acc1 = __builtin_amdgcn_wmma_f32_16x16x4_f32(false, a, false, b1, (short)0, acc1, false, false);
        acc2 = __builtin_amdgcn_wmma_f32_16x16x4_f32(false, a, false, b2, (short)0, acc2, false, false);
        acc3 = __builtin_amdgcn_wmma_f32_16x16x4_f32(false, a, false, b3, (short)0, acc3, false, false);
    }

#pragma unroll
    for (int r = 0; r < 8; ++r) {
        float* o = out + (size_t)(row0 + r + 8 * half) * DIM;
        o[m]      = acc0[r];
        o[16 + m] = acc1[r];
        o[32 + m] = acc2[r];
        o[48 + m] = acc3[r];
    }
}

// ---------------------------------------------------------------------------
// Edge scatter: out[dst,:] += feat[src,:] * dinv[src]*dinv[dst]
// One thread per (edge, feature) element; 64 consecutive lanes share an edge
// -> fully coalesced 256B gather + 256B atomic-add per edge (L2-resident).
// Edges e >= N_EDGES are the N self-loops.
// ---------------------------------------------------------------------------
__global__ void scatter_kernel(const int* __restrict__ src,
                               const int* __restrict__ dst,
                               const float* __restrict__ dinv,
                               const float* __restrict__ feat,
                               float* __restrict__ out) {
    size_t tid = (size_t)blockIdx.x * blockDim.x + threadIdx.x;
    const size_t total = (size_t)(N_EDGES + N_NODES) * DIM;
    if (tid >= total) return;
    const int e = (int)(tid >> 6);
    const int j = (int)(tid & 63);
    int s, d;
    if (e < N_EDGES) { s = src[e]; d = dst[e]; }
    else             { s = e - N_EDGES; d = s; }
    const float w = dinv[s] * dinv[d];
    atomicAdd(out + (size_t)d * DIM + j, feat[(size_t)s * DIM + j] * w);
}

// h = relu(agg1 + b1), in place
__global__ void bias_relu_kernel(float* __restrict__ h, const float* __restrict__ b1) {
    size_t tid = (size_t)blockIdx.x * blockDim.x + threadIdx.x;
    if (tid < (size_t)N_NODES * DIM)
        h[tid] = fmaxf(h[tid] + b1[tid & 63], 0.0f);
}

// ---------------------------------------------------------------------------
// GEMM2: mu = agg2 @ Wmu + bmu ; logvar = agg2 @ Wlv + blv
// (A(hW) == (Ah)W, so we project the aggregated features.)
// One wave per 16-row block; 4 acc tiles: mu[:,0:16], mu[:,16:32],
// lv[:,0:16], lv[:,16:32]. Output: d_out = [mu (N x 32) | logvar (N x 32)].
// ---------------------------------------------------------------------------
__global__ __launch_bounds__(32) void gemm_out(const float* __restrict__ agg,
                                               const float* __restrict__ Wmu,
                                               const float* __restrict__ bmu,
                                               const float* __restrict__ Wlv,
                                               const float* __restrict__ blv,
                                               float* __restrict__ out) {
    const int lane = threadIdx.x;
    const int m    = lane & 15;
    const int half = lane >> 4;
    const int row0 = blockIdx.x * 16;

    v8f accMu0 = {}, accMu1 = {}, accLv0 = {}, accLv1 = {};
    const float* ar = agg + (size_t)(row0 + m) * DIM;

#pragma unroll
    for (int ks = 0; ks < 16; ++ks) {
        const int kk = ks * 4 + half * 2;
        v2f a;  a.x = ar[kk];  a.y = ar[kk + 1];
        const float* mu0 = Wmu + (size_t)kk * OUT_DIM;
        const float* mu1 = Wmu + (size_t)(kk + 1) * OUT_DIM;
        const float* lv0 = Wlv + (size_t)kk * OUT_DIM;
        const float* lv1 = Wlv + (size_t)(kk + 1) * OUT_DIM;
        v2f b0, b1, b2, b3;
        b0.x = mu0[m];      b0.y = mu1[m];
        b1.x = mu0[16 + m]; b1.y = mu1[16 + m];
        b2.x = lv0[m];      b2.y = lv1[m];
        b3.x = lv0[16 + m]; b3.y = lv1[16 + m];
        accMu0 = __builtin_amdgcn_wmma_f32_16x16x4_f32(false, a, false, b0, (short)0, accMu0, false, false);
        accMu1 = __builtin_amdgcn_wmma_f32_16x16x4_f32(false, a, false, b1, (short)0, accMu1, false, false);
        accLv0 = __builtin_amdgcn_wmma_f32_16x16x4_f32(false, a, false, b2, (short)0, accLv0, false, false);
        accLv1 = __builtin_amdgcn_wmma_f32_16x16x4_f32(false, a, false, b3, (short)0, accLv1, false, false);
    }

    float* lvBase = out + (size_t)N_NODES * OUT_DIM;
#pragma unroll
    for (int r = 0; r < 8; ++r) {
        const int row = row0 + r + 8 * half;
        float* omu = out    + (size_t)row * OUT_DIM;
        float* olv = lvBase + (size_t)row * OUT_DIM;
        omu[m]      = accMu0[r] + bmu[m];
        omu[16 + m] = accMu1[r] + bmu[16 + m];
        olv[m]      = accLv0[r] + blv[m];
        olv[16 + m] = accLv1[r] + blv[16 + m];
    }
}

// ---------------------------------------------------------------------------
extern "C" void kernel_launch(void* const* d_in, const int* in_sizes, int n_in,
                              void* d_out, int out_size, void* d_ws, size_t ws_size,
                              hipStream_t stream) {
    const float* x    = (const float*)d_in[0];
    const int*   eidx = (const int*)d_in[1];     // [2, E] flat
    const int*   src  = eidx;
    const int*   dst  = eidx + N_EDGES;
    const float* W1   = (const float*)d_in[2];
    const float* b1   = (const float*)d_in[3];
    const float* Wmu  = (const float*)d_in[4];
    const float* bmu  = (const float*)d_in[5];
    const float* Wlv  = (const float*)d_in[6];
    const float* blv  = (const float*)d_in[7];
    float*       out  = (float*)d_out;

    // workspace: deg/dinv [N] | xw [N*64] | agg1(->h) [N*64] | agg2 [N*64]
    float* deg  = (float*)d_ws;
    float* xw   = deg  + N_NODES;
    float* agg1 = xw   + (size_t)N_NODES * DIM;
    float* agg2 = agg1 + (size_t)N_NODES * DIM;

    const int TB = 256;
    const size_t featElems    = (size_t)N_NODES * DIM;             // 6.4M
    const size_t scatterElems = (size_t)(N_EDGES + N_NODES) * DIM; // 108.8M

    const unsigned gInit    = (unsigned)((featElems + TB - 1) / TB);
    const unsigned gDeg     = (N_EDGES + TB - 1) / TB;
    const unsigned gNode    = (N_NODES + TB - 1) / TB;
    const unsigned gScatter = (unsigned)((scatterElems + TB - 1) / TB);
    const unsigned gGemm    = N_NODES / 16;                        // 6250, exact

    init_kernel<<<gInit, TB, 0, stream>>>(deg, agg1, agg2);
    degree_kernel<<<gDeg, TB, 0, stream>>>(dst, deg);
    rsqrt_kernel<<<gNode, TB, 0, stream>>>(deg);                   // deg -> dinv

    gemm_x_w1<<<gGemm, 32, 0, stream>>>(x, W1, xw);                // xw = x @ W1
    scatter_kernel<<<gScatter, TB, 0, stream>>>(src, dst, deg, xw, agg1);
    bias_relu_kernel<<<gInit, TB, 0, stream>>>(agg1, b1);          // agg1 -> h
    scatter_kernel<<<gScatter, TB, 0, stream>>>(src, dst, deg, agg1, agg2);
    gemm_out<<<gGemm, 32, 0, stream>>>(agg2, Wmu, bmu, Wlv, blv, out);
}